// Cluster_kmeans_pp_23519240913029
// MI455X (gfx1250) — compile-verified
//
#include <hip/hip_runtime.h>

// Problem constants from the reference
#define B_SZ 2048
#define N_SZ 512
#define D_SZ 8192   // C*T = 64*128

typedef float v2f __attribute__((ext_vector_type(2)));
typedef float v8f __attribute__((ext_vector_type(8)));

__device__ __forceinline__ unsigned int order_f32(float f) {
  unsigned int u = __float_as_uint(f);
  return (u & 0x80000000u) ? ~u : (u | 0x80000000u);
}

// ---- kernel 1: mnorm[n] = sum_d m[n,d]^2 ---------------------------------
__global__ void k_mnorm(const float* __restrict__ m, float* __restrict__ mnorm) {
  const int n = blockIdx.x;
  const float* row = m + (size_t)n * D_SZ;
  float s = 0.f;
  for (int d = threadIdx.x; d < D_SZ; d += 256) { float v = row[d]; s += v * v; }
  __shared__ float red[256];
  red[threadIdx.x] = s;
  __syncthreads();
  for (int off = 128; off > 0; off >>= 1) {
    if ((int)threadIdx.x < off) red[threadIdx.x] += red[threadIdx.x + off];
    __syncthreads();
  }
  if (threadIdx.x == 0) mnorm[n] = red[0];
}

// ---- kernel 2: init argmin keys ------------------------------------------
__global__ void k_init_keys(unsigned long long* keys) {
  int i = blockIdx.x * 256 + threadIdx.x;
  if (i < B_SZ) keys[i] = 0xFFFFFFFFFFFFFFFFull;
}

// ---- pipeline helpers: constant-index only (guarantees SSA promotion) ----
__device__ __forceinline__ void load_chunk(const float* __restrict__ yrow,
                                           const float* __restrict__ m0,
                                           const float* __restrict__ m1,
                                           const float* __restrict__ m2,
                                           const float* __restrict__ m3,
                                           int k, v2f a[2], v2f b[2][4]) {
  a[0]    = *(const v2f*)(yrow + k);
  b[0][0] = *(const v2f*)(m0 + k);
  b[0][1] = *(const v2f*)(m1 + k);
  b[0][2] = *(const v2f*)(m2 + k);
  b[0][3] = *(const v2f*)(m3 + k);
  a[1]    = *(const v2f*)(yrow + k + 4);
  b[1][0] = *(const v2f*)(m0 + k + 4);
  b[1][1] = *(const v2f*)(m1 + k + 4);
  b[1][2] = *(const v2f*)(m2 + k + 4);
  b[1][3] = *(const v2f*)(m3 + k + 4);
}

__device__ __forceinline__ void comp_chunk(const v2f a[2], const v2f b[2][4],
                                           v8f acc[4]) {
#pragma unroll
  for (int s = 0; s < 2; ++s) {
#pragma unroll
    for (int j = 0; j < 4; ++j)
      acc[j] = __builtin_amdgcn_wmma_f32_16x16x4_f32(
          false, a[s], false, b[s][j], (short)0, acc[j], false, false);
  }
}

// ---- kernel 3: WMMA fp32 distance GEMM + running argmin ------------------
// grid = B/16 row tiles, 4 waves/block.
// Wave w owns column tiles [8w, 8w+8), processed as 2 passes of 4 tiles.
// Per pass: 4 accumulators share one A fragment; K loop software-pipelined
// with a statically unrolled 2-stage ping-pong (buffers are named variables,
// never runtime-indexed).
__global__ void __launch_bounds__(128)
k_assign(const float* __restrict__ y, const float* __restrict__ m,
         const float* __restrict__ mnorm, unsigned long long* __restrict__ keys) {
  const int lane = threadIdx.x & 31;
  const int wave = threadIdx.x >> 5;     // 0..3
  const int rowTile = blockIdx.x;        // 0..127
  const int r16 = lane & 15;
  const int hl  = lane >> 4;             // 0 or 1

  // fp32 WMMA A-fragment addressing: lane holds [row = lane%16][K = 2*hl, 2*hl+1]
  const float* yrow = y + (size_t)(rowTile * 16 + r16) * D_SZ + 2 * hl;

  float minval[8];
  int   minidx[8];
#pragma unroll
  for (int v = 0; v < 8; ++v) { minval[v] = 3.4e38f; minidx[v] = 0; }

  const int NCH = D_SZ / 8;              // 1024 chunks of 2 K-steps (8 K each)

  for (int pass = 0; pass < 2; ++pass) {
    const int ct0 = wave * 8 + pass * 4; // first of 4 column tiles this pass

    const float* m0 = m + (size_t)((ct0 + 0) * 16 + r16) * D_SZ + 2 * hl;
    const float* m1 = m + (size_t)((ct0 + 1) * 16 + r16) * D_SZ + 2 * hl;
    const float* m2 = m + (size_t)((ct0 + 2) * 16 + r16) * D_SZ + 2 * hl;
    const float* m3 = m + (size_t)((ct0 + 3) * 16 + r16) * D_SZ + 2 * hl;

    v8f acc[4];
#pragma unroll
    for (int j = 0; j < 4; ++j) acc[j] = (v8f){0.f,0.f,0.f,0.f,0.f,0.f,0.f,0.f};

    // two statically named pipeline stages
    v2f a0[2], b0[2][4];
    v2f a1[2], b1[2][4];

    load_chunk(yrow, m0, m1, m2, m3, 0, a0, b0);          // chunk 0 -> buf0

    for (int c = 0; c < NCH - 2; c += 2) {
      load_chunk(yrow, m0, m1, m2, m3, (c + 1) * 8, a1, b1); // chunk c+1 -> buf1
      comp_chunk(a0, b0, acc);                               // compute chunk c
      load_chunk(yrow, m0, m1, m2, m3, (c + 2) * 8, a0, b0); // chunk c+2 -> buf0
      comp_chunk(a1, b1, acc);                               // compute chunk c+1
    }
    // epilogue: buf0 holds chunk NCH-2
    load_chunk(yrow, m0, m1, m2, m3, (NCH - 1) * 8, a1, b1);
    comp_chunk(a0, b0, acc);
    comp_chunk(a1, b1, acc);

    // fold 4 tiles into the running per-lane (min, argmin)
#pragma unroll
    for (int j = 0; j < 4; ++j) {
      const int col = (ct0 + j) * 16 + r16;
      const float mn = mnorm[col];   // ||m_col||^2 ; ||y||^2 dropped (constant per row)
#pragma unroll
      for (int v = 0; v < 8; ++v) {
        float s = mn - 2.0f * acc[j][v];
        if (s < minval[v]) { minval[v] = s; minidx[v] = col; }
      }
    }
  }

  // Reduce (min, argmin) within each 16-lane group (rows differ per half/VGPR)
#pragma unroll
  for (int v = 0; v < 8; ++v) {
    float mv = minval[v];
    int   mi = minidx[v];
#pragma unroll
    for (int off = 8; off >= 1; off >>= 1) {
      float ov = __shfl_xor(mv, off, 32);
      int   oi = __shfl_xor(mi, off, 32);
      if (ov < mv || (ov == mv && oi < mi)) { mv = ov; mi = oi; }
    }
    if (r16 == 0) {
      int row = rowTile * 16 + v + 8 * hl;  // C/D layout: VGPR v -> rows v, v+8
      unsigned long long key =
          ((unsigned long long)order_f32(mv) << 32) | (unsigned int)mi;
      atomicMin(&keys[row], key);           // global_atomic_min_u64 cross-wave combine
    }
  }
}

// ---- kernel 4: decode keys -> assignment (int for step 5, float for output)
__global__ void k_extract(const unsigned long long* __restrict__ keys,
                          int* __restrict__ assign_i, float* __restrict__ assign_f) {
  int i = blockIdx.x * 256 + threadIdx.x;
  if (i < B_SZ) {
    int a = (int)(keys[i] & 0xFFFFFFFFull);
    assign_i[i] = a;
    assign_f[i] = (float)a;
  }
}

// ---- kernel 5: per-centroid order-faithful EMA updates -------------------
// block n owns centroid row n; row held in registers (32 m + 32 sd per thread)
__global__ void __launch_bounds__(256)
k_ema(const float* __restrict__ y, const float* __restrict__ m_in,
      const float* __restrict__ sd_in, const float* __restrict__ p_in,
      const int* __restrict__ assign,
      float* __restrict__ m_out, float* __restrict__ sd_out,
      float* __restrict__ p_out) {
  const int n = blockIdx.x;
  const int t = threadIdx.x;
  const int PER = D_SZ / 256;   // 32
  float mreg[32], sreg[32];
  const size_t base = (size_t)n * D_SZ;
#pragma unroll
  for (int j = 0; j < PER; ++j) {
    int d = j * 256 + t;
    mreg[j] = m_in[base + d];
    sreg[j] = sd_in[base + d];
  }
  int count = 0;
  for (int b = 0; b < B_SZ; ++b) {
    if (assign[b] == n) {       // uniform branch per block, ordered in b
      ++count;
      const float* yr = y + (size_t)b * D_SZ;
#pragma unroll
      for (int j = 0; j < PER; ++j) {
        int d = j * 256 + t;
        float yv = yr[d];
        float mi = fmaf(mreg[j], 0.001f, yv * 0.999f);
        mreg[j] = mi;
        float diff = mi - yv;
        sreg[j] = fmaf(diff * diff, 0.001f, sreg[j] * 0.999f);
      }
    }
  }
#pragma unroll
  for (int j = 0; j < PER; ++j) {
    int d = j * 256 + t;
    m_out[base + d]  = mreg[j];
    sd_out[base + d] = sreg[j];
  }
  if (t == 0) p_out[n] = p_in[n] + (float)count;
}

// ---- launcher -------------------------------------------------------------
extern "C" void kernel_launch(void* const* d_in, const int* in_sizes, int n_in,
                              void* d_out, int out_size, void* d_ws, size_t ws_size,
                              hipStream_t stream) {
  const float* y    = (const float*)d_in[0];   // [B, C, T]
  const float* m    = (const float*)d_in[1];   // [N, C, T]
  const float* sd   = (const float*)d_in[2];   // [N, C, T]
  const float* p    = (const float*)d_in[3];   // [N]

  // workspace layout
  unsigned long long* keys = (unsigned long long*)d_ws;                        // B * 8
  float* mnorm    = (float*)((char*)d_ws + B_SZ * sizeof(unsigned long long)); // N * 4
  int*   assign_i = (int*)((char*)d_ws + B_SZ * 8 + N_SZ * 4);                 // B * 4

  // output layout: m | sd | p | assign (all float32)
  float* m_out   = (float*)d_out;
  float* sd_out  = m_out + (size_t)N_SZ * D_SZ;
  float* p_out   = sd_out + (size_t)N_SZ * D_SZ;
  float* asn_out = p_out + N_SZ;

  k_mnorm<<<N_SZ, 256, 0, stream>>>(m, mnorm);
  k_init_keys<<<(B_SZ + 255) / 256, 256, 0, stream>>>(keys);
  k_assign<<<B_SZ / 16, 128, 0, stream>>>(y, m, mnorm, keys);
  k_extract<<<(B_SZ + 255) / 256, 256, 0, stream>>>(keys, assign_i, asn_out);
  k_ema<<<N_SZ, 256, 0, stream>>>(y, m, sd, p, assign_i, m_out, sd_out, p_out);
}